// Net_78091095376068
// MI455X (gfx1250) — compile-verified
//
#include <hip/hip_runtime.h>

typedef __attribute__((ext_vector_type(2))) float v2f;
typedef __attribute__((ext_vector_type(4))) float v4f;
typedef __attribute__((ext_vector_type(8))) float v8f;

#define TSTEPS 100
#define BATCH  8192
#define H1     300
#define H1P    304      // K padded to multiple of 4 (and of 16)
#define NOUT   10
#define KIN    784
#define ROWS   64       // batch rows per workgroup
#define NTILES 19       // 304/16 column tiles of cur1

__global__ __launch_bounds__(256, 1) void snn_fused_kernel(
    const float* __restrict__ x,  const float* __restrict__ W1,
    const float* __restrict__ b1, const float* __restrict__ W2,
    const float* __restrict__ b2, const float* __restrict__ beta2,
    float* __restrict__ outSpk,   float* __restrict__ outMem)
{
    __shared__ __align__(16) float cur1_s[ROWS * H1P];   // 77,824 B
    __shared__ __align__(16) float mem1_s[ROWS * H1P];   // 77,824 B
    __shared__ __align__(16) float w2B[(H1P / 2) * 32];  // 19,456 B, B-fragment order
    __shared__ float b2_s[16];
    __shared__ float bet_s[16];

    const int tid  = threadIdx.x;
    const int wv   = tid >> 5;          // wave id (wave32)
    const int lane = tid & 31;
    const int half = lane >> 4;         // lane half selects K pair
    const int l16  = lane & 15;         // N (B/C/D) or M (A) index
    const int rowBase = blockIdx.x * ROWS;

    // ---- W2 into B-fragment order: pair (k, k+1) contiguous per column n ----
    for (int idx = tid; idx < H1P * 16; idx += 256) {
        int k = idx >> 4, nn = idx & 15;
        float v = (k < H1 && nn < NOUT) ? W2[nn * H1 + k] : 0.0f;
        w2B[(k >> 1) * 32 + nn * 2 + (k & 1)] = v;
    }
    if (tid < 16) {
        b2_s[tid] = (tid < NOUT) ? b2[tid] : 0.0f;
        float bb = (tid < NOUT) ? beta2[tid] : 0.0f;
        bet_s[tid] = fminf(fmaxf(bb, 0.0f), 1.0f);   // snntorch beta clamp
    }
    {   // zero mem1 (vectorized)
        v4f z = {0.f, 0.f, 0.f, 0.f};
        v4f* m4 = (v4f*)mem1_s;
        for (int idx = tid; idx < (ROWS * H1P) / 4; idx += 256) m4[idx] = z;
    }

    // ---- one-time GEMM: cur1 = x @ W1^T + b1 for this block's 64 rows ----
    // 4 row-tiles x 19 col-tiles = 76 wave-jobs over 8 waves, f32 WMMA 16x16x4
    for (int job = wv; job < (ROWS / 16) * NTILES; job += 8) {
        int rt = job / NTILES, nt = job % NTILES;
        int col  = nt * 16 + l16;
        int colc = (col < H1) ? col : (H1 - 1);     // clamp to avoid OOB reads
        const float* aP = x  + (size_t)(rowBase + rt * 16 + l16) * KIN;
        const float* bP = W1 + (size_t)colc * KIN;
        float bias = (col < H1) ? b1[col] : 0.0f;
        v8f c;
        #pragma unroll
        for (int j = 0; j < 8; ++j) c[j] = bias;
        #pragma unroll 4
        for (int k0 = 0; k0 < KIN; k0 += 4) {
            v2f a = *(const v2f*)(aP + k0 + 2 * half);   // global_load_b64
            v2f b = *(const v2f*)(bP + k0 + 2 * half);   // global_load_b64
            c = __builtin_amdgcn_wmma_f32_16x16x4_f32(false, a, false, b,
                                                      (short)0, c, false, false);
        }
        #pragma unroll
        for (int j = 0; j < 8; ++j) {
            int rr = rt * 16 + j + 8 * half;
            cur1_s[rr * H1P + col] = (col < H1) ? c[j] : 0.0f;  // pad cols -> 0
        }
    }
    __syncthreads();

    // mem2 accumulator lives in registers in C/D fragment layout (waves 0..3)
    v8f m2 = {0.f,0.f,0.f,0.f,0.f,0.f,0.f,0.f};
    float bias2 = 0.0f, beta = 0.0f;
    const float* aMem = nullptr;
    if (wv < 4) {
        bias2 = b2_s[l16];
        beta  = bet_s[l16];
        aMem  = &mem1_s[(wv * 16 + l16) * H1P];
    }

    const v4f* cur4 = (const v4f*)cur1_s;
    v4f*       mm4  = (v4f*)mem1_s;

    for (int t = 0; t < TSTEPS; ++t) {
        // -- phase 1: mem1 leaky-integrate, float4-wide (19 iters/thread) --
        #pragma unroll 4
        for (int idx = tid; idx < (ROWS * H1P) / 4; idx += 256) {
            v4f m = mm4[idx];               // ds_load_b128
            v4f cu = cur4[idx];             // ds_load_b128
            v4f r;
            #pragma unroll
            for (int e = 0; e < 4; ++e) {
                float mo = m[e];
                float r1 = (mo > 1.0f) ? 1.0f : 0.0f;        // reset from prev mem
                r[e] = fmaf(0.9f, mo, cu[e]) - r1;           // beta1*m + cur1 - reset
            }
            mm4[idx] = r;                   // ds_store_b128
        }
        __syncthreads();

        // -- phase 2: cur2 = spike(mem1) @ W2^T + b2 via f32 WMMA; mem2 update --
        if (wv < 4) {                       // wave-uniform branch: EXEC all-1s
            v8f c;
            #pragma unroll
            for (int j = 0; j < 8; ++j) c[j] = bias2;
            #pragma unroll 4
            for (int k0 = 0; k0 < H1P; k0 += 4) {
                v2f mv = *(const v2f*)(aMem + k0 + 2 * half);    // ds_load_b64
                v2f a;
                a.x = (mv.x > 1.0f) ? 1.0f : 0.0f;               // spike on the fly
                a.y = (mv.y > 1.0f) ? 1.0f : 0.0f;
                v2f b = *(const v2f*)(&w2B[((k0 >> 1) + half) * 32 + l16 * 2]);
                c = __builtin_amdgcn_wmma_f32_16x16x4_f32(false, a, false, b,
                                                          (short)0, c, false, false);
            }
            #pragma unroll
            for (int j = 0; j < 8; ++j) {
                float mo = m2[j];
                float r2 = (mo > 1.0f) ? 1.0f : 0.0f;
                float mn = fmaf(beta, mo, c[j]) - r2;
                m2[j] = mn;
                if (l16 < NOUT) {
                    size_t o = (size_t)t * (BATCH * NOUT)
                             + (size_t)(rowBase + wv * 16 + j + 8 * half) * NOUT + l16;
                    outSpk[o] = (mn > 1.0f) ? 1.0f : 0.0f;
                    outMem[o] = mn;
                }
            }
        }
        __syncthreads();
    }
}

extern "C" void kernel_launch(void* const* d_in, const int* in_sizes, int n_in,
                              void* d_out, int out_size, void* d_ws, size_t ws_size,
                              hipStream_t stream) {
    (void)in_sizes; (void)n_in; (void)d_ws; (void)ws_size; (void)out_size;
    const float* x   = (const float*)d_in[0];
    const float* W1  = (const float*)d_in[1];
    const float* b1  = (const float*)d_in[2];
    const float* W2  = (const float*)d_in[3];
    const float* b2  = (const float*)d_in[4];
    const float* bt  = (const float*)d_in[5];
    float* outSpk = (float*)d_out;
    float* outMem = outSpk + (size_t)TSTEPS * BATCH * NOUT;
    snn_fused_kernel<<<dim3(BATCH / ROWS), dim3(256), 0, stream>>>(
        x, W1, b1, W2, b2, bt, outSpk, outMem);
}